// PGCLEncoder_51187420233788
// MI455X (gfx1250) — compile-verified
//
#include <hip/hip_runtime.h>
#include <hip/hip_bf16.h>

typedef float v2f __attribute__((ext_vector_type(2)));
typedef float v8f __attribute__((ext_vector_type(8)));

#define WMMA_F32(a, b, c) \
    __builtin_amdgcn_wmma_f32_16x16x4_f32(false, (a), false, (b), (short)0, (c), false, false)

#define DIM 64
#define OUT_DIM 192
#define NGRAPH 512
#define BN_EPS 1e-5f
// padded LDS row strides (floats): stride % 64 == 4 -> bank = 4*l16 + k, two
// lane-halves offset by 2 banks -> conflict-free; stride*4 % 16 == 0 -> b128-aligned rows
#define TS_NODE 68
#define TS_POOL 196

// ---------------------------------------------------------------- utilities
__global__ void zero_f32_kernel(float* __restrict__ p, int n) {
    int i = blockIdx.x * blockDim.x + threadIdx.x;
    if (i < n) p[i] = 0.0f;
}

__global__ void copy4_kernel(float4* __restrict__ dst, const float4* __restrict__ src, int n4) {
    int i = blockIdx.x * blockDim.x + threadIdx.x;
    if (i < n4) dst[i] = src[i];
}

// ------------------------------------------------------- edge scatter (GIN agg)
// one thread per (edge, 4 consecutive dims): agg[dst] += h[src]
__global__ void edge_scatter_kernel(float* __restrict__ agg, const float* __restrict__ h,
                                    const int* __restrict__ ei, int nE) {
    int gid = blockIdx.x * blockDim.x + threadIdx.x;
    int total = nE * 16;
    if (gid >= total) return;
    int e = gid >> 4;
    int c = (gid & 15) << 2;
    int s = ei[e];
    int d = ei[nE + e];
    float4 v = *(const float4*)(h + (size_t)s * DIM + c);
    float* base = agg + (size_t)d * DIM + c;
    atomicAdd(base + 0, v.x);
    atomicAdd(base + 1, v.y);
    atomicAdd(base + 2, v.z);
    atomicAdd(base + 3, v.w);
}

// ------------------------------------------------- fused node MLP (GIN update)
// hout = relu( relu(agg @ W1 + b1) @ W2 + b2 ), per-wave 16-row tile.
// agg tile staged to LDS (coalesced b128), f32 WMMA 16x16x4, padded LDS strides.
__global__ __launch_bounds__(128) void node_mlp_kernel(
    float* __restrict__ hout, const float* __restrict__ agg,
    const float* __restrict__ W1, const float* __restrict__ b1,
    const float* __restrict__ W2, const float* __restrict__ b2, int N) {
    __shared__ float lds[4 * 16 * TS_NODE];
    const int wave = threadIdx.x >> 5;
    const int lane = threadIdx.x & 31;
    const int half = lane >> 4;
    const int l16  = lane & 15;
    const int row0 = (blockIdx.x * 4 + wave) * 16;
    float* tile = lds + wave * 16 * TS_NODE;

    // stage agg 16x64 tile into LDS, coalesced (16 rows x 16 float4)
#pragma unroll
    for (int i = 0; i < 8; ++i) {
        int idx  = i * 32 + lane;   // 0..255
        int row  = idx >> 4;
        int quad = idx & 15;
        int gr = row0 + row;
        if (gr >= N) gr = N - 1;    // clamp tail (masked at final store)
        float4 v = *(const float4*)(agg + (size_t)gr * DIM + quad * 4);
        *(float4*)(tile + row * TS_NODE + quad * 4) = v;
    }
    __syncthreads();

    // GEMM1: Z = agg_tile(16x64) @ W1(64x64)
    v8f acc[4] = {};
    for (int k0 = 0; k0 < DIM; k0 += 4) {
        v2f a;
        a.x = tile[l16 * TS_NODE + k0 + 2 * half];
        a.y = tile[l16 * TS_NODE + k0 + 2 * half + 1];
#pragma unroll
        for (int ct = 0; ct < 4; ++ct) {
            int col = ct * 16 + l16;
            v2f b;
            b.x = W1[(k0 + 2 * half) * DIM + col];
            b.y = W1[(k0 + 2 * half + 1) * DIM + col];
            acc[ct] = WMMA_F32(a, b, acc[ct]);
        }
    }
    __syncthreads();
    // bias + relu -> LDS (overwrite tile; D-layout -> row-major)
#pragma unroll
    for (int ct = 0; ct < 4; ++ct) {
        int col = ct * 16 + l16;
        float bias = b1[col];
#pragma unroll
        for (int v = 0; v < 8; ++v) {
            float val = acc[ct][v] + bias;
            val = val > 0.0f ? val : 0.0f;
            tile[(v + 8 * half) * TS_NODE + col] = val;
        }
    }
    __syncthreads();
    // GEMM2: H = Zr(16x64) @ W2(64x64)
    v8f acc2[4] = {};
    for (int k0 = 0; k0 < DIM; k0 += 4) {
        v2f a;
        a.x = tile[l16 * TS_NODE + k0 + 2 * half];
        a.y = tile[l16 * TS_NODE + k0 + 2 * half + 1];
#pragma unroll
        for (int ct = 0; ct < 4; ++ct) {
            int col = ct * 16 + l16;
            v2f b;
            b.x = W2[(k0 + 2 * half) * DIM + col];
            b.y = W2[(k0 + 2 * half + 1) * DIM + col];
            acc2[ct] = WMMA_F32(a, b, acc2[ct]);
        }
    }
#pragma unroll
    for (int ct = 0; ct < 4; ++ct) {
        int col = ct * 16 + l16;
        float bias = b2[col];
#pragma unroll
        for (int v = 0; v < 8; ++v) {
            float val = acc2[ct][v] + bias;
            val = val > 0.0f ? val : 0.0f;
            int rr = row0 + v + 8 * half;
            if (rr < N) hout[(size_t)rr * DIM + col] = val;
        }
    }
}

// ----------------------------------------------------------- BN column stats
__global__ __launch_bounds__(256) void col_stats_kernel(float* __restrict__ stats,
                                                        const float* __restrict__ h,
                                                        int N, int rowsPerBlock) {
    __shared__ float ss[256];
    __shared__ float sq[256];
    int col = threadIdx.x & 63;
    int rg  = threadIdx.x >> 6;
    float s = 0.0f, s2 = 0.0f;
    int r0 = blockIdx.x * rowsPerBlock;
    int r1 = r0 + rowsPerBlock;
    if (r1 > N) r1 = N;
    for (int r = r0 + rg; r < r1; r += 4) {
        float v = h[(size_t)r * DIM + col];
        s += v;
        s2 += v * v;
    }
    ss[threadIdx.x] = s;
    sq[threadIdx.x] = s2;
    __syncthreads();
    if (threadIdx.x < 64) {
        float ts = ss[threadIdx.x] + ss[threadIdx.x + 64] + ss[threadIdx.x + 128] + ss[threadIdx.x + 192];
        float tq = sq[threadIdx.x] + sq[threadIdx.x + 64] + sq[threadIdx.x + 128] + sq[threadIdx.x + 192];
        atomicAdd(stats + col, ts);
        atomicAdd(stats + 64 + col, tq);
    }
}

// ------------------------------------- BN apply (in place) + segment-sum pool
// batch is sorted: run-length aggregate per 128-node chunk, one atomic per
// (graph-run, column) instead of one per (node, column).
__global__ __launch_bounds__(64) void bn_pool_kernel(float* __restrict__ h,
                                                     const float* __restrict__ stats,
                                                     const float* __restrict__ gamma,
                                                     const float* __restrict__ beta,
                                                     const int* __restrict__ batch,
                                                     float* __restrict__ pooled,
                                                     int layer, int N, int chunk) {
    int col = threadIdx.x;  // 0..63
    int n0 = blockIdx.x * chunk;
    int n1 = n0 + chunk;
    if (n1 > N) n1 = N;
    if (n0 >= N) return;

    float invN = 1.0f / (float)N;
    float mean = stats[col] * invN;
    float var  = stats[64 + col] * invN - mean * mean;
    float sc   = gamma[col] * rsqrtf(var + BN_EPS);
    float bt   = beta[col];

    float acc = 0.0f;
    int gcur = batch[n0];
    for (int n = n0; n < n1; ++n) {
        int g = batch[n];               // uniform across block
        if (g != gcur) {
            atomicAdd(&pooled[gcur * OUT_DIM + layer * DIM + col], acc);
            acc = 0.0f;
            gcur = g;
        }
        size_t idx = (size_t)n * DIM + col;
        float val = (h[idx] - mean) * sc + bt;
        h[idx] = val;
        acc += val;
    }
    atomicAdd(&pooled[gcur * OUT_DIM + layer * DIM + col], acc);
}

// ------------------------------------------------ pooled projection head MLP
// y = relu(pooled @ pW1 + pb1) @ pW2 + pb2, G=512 rows, 192x192 weights.
__global__ __launch_bounds__(64) void pooled_mlp_kernel(
    float* __restrict__ y, const float* __restrict__ pooled,
    const float* __restrict__ pW1, const float* __restrict__ pb1,
    const float* __restrict__ pW2, const float* __restrict__ pb2) {
    __shared__ float lds[2 * 16 * TS_POOL];
    const int wave = threadIdx.x >> 5;
    const int lane = threadIdx.x & 31;
    const int half = lane >> 4;
    const int l16  = lane & 15;
    const int row0 = (blockIdx.x * 2 + wave) * 16;
    float* tile = lds + wave * 16 * TS_POOL;

    // stage pooled 16x192 tile into LDS, coalesced (16 rows x 48 float4)
#pragma unroll
    for (int i = 0; i < 24; ++i) {
        int idx  = i * 32 + lane;   // 0..767
        int row  = idx / 48;
        int quad = idx % 48;
        float4 v = *(const float4*)(pooled + (row0 + row) * OUT_DIM + quad * 4);
        *(float4*)(tile + row * TS_POOL + quad * 4) = v;
    }
    __syncthreads();

    v8f acc[12] = {};
    for (int k0 = 0; k0 < OUT_DIM; k0 += 4) {
        v2f a;
        a.x = tile[l16 * TS_POOL + k0 + 2 * half];
        a.y = tile[l16 * TS_POOL + k0 + 2 * half + 1];
#pragma unroll
        for (int ct = 0; ct < 12; ++ct) {
            int col = ct * 16 + l16;
            v2f b;
            b.x = pW1[(k0 + 2 * half) * OUT_DIM + col];
            b.y = pW1[(k0 + 2 * half + 1) * OUT_DIM + col];
            acc[ct] = WMMA_F32(a, b, acc[ct]);
        }
    }
    __syncthreads();
#pragma unroll
    for (int ct = 0; ct < 12; ++ct) {
        int col = ct * 16 + l16;
        float bias = pb1[col];
#pragma unroll
        for (int v = 0; v < 8; ++v) {
            float val = acc[ct][v] + bias;
            val = val > 0.0f ? val : 0.0f;
            tile[(v + 8 * half) * TS_POOL + col] = val;
        }
    }
    __syncthreads();
    v8f acc2[12] = {};
    for (int k0 = 0; k0 < OUT_DIM; k0 += 4) {
        v2f a;
        a.x = tile[l16 * TS_POOL + k0 + 2 * half];
        a.y = tile[l16 * TS_POOL + k0 + 2 * half + 1];
#pragma unroll
        for (int ct = 0; ct < 12; ++ct) {
            int col = ct * 16 + l16;
            v2f b;
            b.x = pW2[(k0 + 2 * half) * OUT_DIM + col];
            b.y = pW2[(k0 + 2 * half + 1) * OUT_DIM + col];
            acc2[ct] = WMMA_F32(a, b, acc2[ct]);
        }
    }
#pragma unroll
    for (int ct = 0; ct < 12; ++ct) {
        int col = ct * 16 + l16;
        float bias = pb2[col];
#pragma unroll
        for (int v = 0; v < 8; ++v) {
            y[(row0 + v + 8 * half) * OUT_DIM + col] = acc2[ct][v] + bias;
        }
    }
}

// ------------------------------------------------- final row L2 normalization
// d_out = [ yn (512x192) | xn (512x192) ]
__global__ __launch_bounds__(192) void normalize_out_kernel(float* __restrict__ out,
                                                            const float* __restrict__ pooled,
                                                            const float* __restrict__ y) {
    __shared__ float sp[192];
    __shared__ float sy[192];
    int g = blockIdx.x;
    int t = threadIdx.x;
    float pv = pooled[g * OUT_DIM + t];
    float yv = y[g * OUT_DIM + t];
    sp[t] = pv * pv;
    sy[t] = yv * yv;
    __syncthreads();
    if (t < 64) {
        sp[t] += sp[t + 64] + sp[t + 128];
        sy[t] += sy[t + 64] + sy[t + 128];
    }
    __syncthreads();
    if (t < 16) {
        float a = 0.0f, b = 0.0f;
        for (int i = t; i < 64; i += 16) { a += sp[i]; b += sy[i]; }
        sp[t] = a;
        sy[t] = b;
    }
    __syncthreads();
    if (t == 0) {
        float a = 0.0f, b = 0.0f;
        for (int i = 0; i < 16; ++i) { a += sp[i]; b += sy[i]; }
        sp[0] = sqrtf(a);
        sy[0] = sqrtf(b);
    }
    __syncthreads();
    float np = fmaxf(sp[0], 1e-12f);
    float ny = fmaxf(sy[0], 1e-12f);
    out[g * OUT_DIM + t] = yv / ny;                          // yn
    out[NGRAPH * OUT_DIM + g * OUT_DIM + t] = pv / np;       // xn
}

// --------------------------------------------------------------------- launch
extern "C" void kernel_launch(void* const* d_in, const int* in_sizes, int n_in,
                              void* d_out, int out_size, void* d_ws, size_t ws_size,
                              hipStream_t stream) {
    const float* x      = (const float*)d_in[0];
    const int*   ei     = (const int*)d_in[1];
    const int*   batch  = (const int*)d_in[2];
    const float* W1s    = (const float*)d_in[3];
    const float* b1s    = (const float*)d_in[4];
    const float* W2s    = (const float*)d_in[5];
    const float* b2s    = (const float*)d_in[6];
    const float* gammas = (const float*)d_in[7];
    const float* betas  = (const float*)d_in[8];
    const float* pW1    = (const float*)d_in[9];
    const float* pb1    = (const float*)d_in[10];
    const float* pW2    = (const float*)d_in[11];
    const float* pb2    = (const float*)d_in[12];
    float* out = (float*)d_out;

    const int N  = in_sizes[0] / DIM;        // 100000
    const int nE = in_sizes[1] / 2;          // 1250000
    const size_t nodeElems = (size_t)N * DIM;

    float* bufA   = (float*)d_ws;            // agg buffer
    float* bufB   = bufA + nodeElems;
    float* bufC   = bufB + nodeElems;
    float* pooled = bufC + nodeElems;                         // [512,192]
    float* ybuf   = pooled + NGRAPH * OUT_DIM;                // [512,192]
    float* stats  = ybuf + NGRAPH * OUT_DIM;                  // [128]

    // zero pooled accumulator
    {
        int n = NGRAPH * OUT_DIM;
        zero_f32_kernel<<<(n + 255) / 256, 256, 0, stream>>>(pooled, n);
    }

    const float* hin[3]  = { x, bufB, bufC };
    float*       hout[3] = { bufB, bufC, bufB };

    const int n4 = N * (DIM / 4);
    const int edgeThreads = nE * 16;
    const int statsRows = 250;
    const int statsBlocks = (N + statsRows - 1) / statsRows;
    const int poolChunk = 128;
    const int poolBlocks = (N + poolChunk - 1) / poolChunk;

    for (int l = 0; l < 3; ++l) {
        const float* W1 = W1s + l * DIM * DIM;
        const float* b1 = b1s + l * DIM;
        const float* W2 = W2s + l * DIM * DIM;
        const float* b2 = b2s + l * DIM;

        // agg = h (self term, eps=0)
        copy4_kernel<<<(n4 + 255) / 256, 256, 0, stream>>>((float4*)bufA, (const float4*)hin[l], n4);
        // agg[dst] += h[src]
        edge_scatter_kernel<<<(edgeThreads + 255) / 256, 256, 0, stream>>>(bufA, hin[l], ei, nE);
        // fused 2-layer MLP with WMMA
        node_mlp_kernel<<<(N + 63) / 64, 128, 0, stream>>>(hout[l], bufA, W1, b1, W2, b2, N);
        // BN stats
        zero_f32_kernel<<<1, 128, 0, stream>>>(stats, 128);
        col_stats_kernel<<<statsBlocks, 256, 0, stream>>>(stats, hout[l], N, statsRows);
        // BN apply + run-length segment pooling (batch is sorted)
        bn_pool_kernel<<<poolBlocks, 64, 0, stream>>>(hout[l], stats,
                                                      gammas + l * DIM, betas + l * DIM,
                                                      batch, pooled, l, N, poolChunk);
    }

    // projection head (WMMA) + row-normalized outputs
    pooled_mlp_kernel<<<NGRAPH / 32, 64, 0, stream>>>(ybuf, pooled, pW1, pb1, pW2, pb2);
    normalize_out_kernel<<<NGRAPH, 192, 0, stream>>>(out, pooled, ybuf);
}